// RNN_4655744549058
// MI455X (gfx1250) — compile-verified
//
#include <hip/hip_runtime.h>
#include <math.h>

// ---- problem constants (from reference) ----
#define NTRIALS 32
#define TSTEPS  1000
#define NN      1024
#define INDIM   64
#define OUTDIM  32
#define DT      0.1f
#define SQRTDT  0.31622776601683794f
#define T1      (TSTEPS + 1)
#define XSEQ_TRIAL_STRIDE (T1 * NN)              // 1025024
#define XSEQ_TOTAL ((long)NTRIALS * XSEQ_TRIAL_STRIDE)
#define Y_TRIAL_STRIDE (TSTEPS * OUTDIM)         // 32000

typedef float v2f __attribute__((ext_vector_type(2)));
typedef float v8f __attribute__((ext_vector_type(8)));

__device__ __forceinline__ float sigmoidf_(float x) {
  return 1.0f / (1.0f + __expf(-x));
}

// D = A(16x4) * B(4x16) + C, fp32 WMMA (wave32)
__device__ __forceinline__ v8f wmma_f32_k4(v2f a, v2f b, v8f c) {
  return __builtin_amdgcn_wmma_f32_16x16x4_f32(
      /*neg_a=*/false, a, /*neg_b=*/false, b,
      /*c_mod=*/(short)0, c, /*reuse_a=*/false, /*reuse_b=*/false);
}

// ---------------------------------------------------------------------------
// Init: transposes (J->Jt[k][n], B->Bt[k][n], W->Wt[k][o]), c[n]=sqrt(dt)*sig^2,
//       s0 = sigmoid(0) = 0.5, x_seq[:,0,:] = 0.
// Launched with NN*NN threads (4096 blocks x 256).
// ---------------------------------------------------------------------------
__global__ __launch_bounds__(256) void rnn_init_kernel(
    const float* __restrict__ Bm, const float* __restrict__ J,
    const float* __restrict__ W, const float* __restrict__ sig,
    float* __restrict__ Jt, float* __restrict__ Bt, float* __restrict__ Wt,
    float* __restrict__ cvec, float* __restrict__ s0, float* __restrict__ out) {
  int idx = blockIdx.x * blockDim.x + threadIdx.x;   // 0 .. 1M-1
  int k = idx >> 10, n = idx & 1023;
  Jt[idx] = J[n * NN + k];                           // Jt[k][n] = J[n][k]
  if (k < INDIM)        Bt[idx] = Bm[n * INDIM + k]; // Bt[k][n] = B[n][k]
  if (idx < NN * OUTDIM) {
    int kk = idx >> 5, o = idx & 31;
    Wt[idx] = W[o * NN + kk];                        // Wt[k][o] = W[o][k]
  }
  if (idx < NN) { float s = sig[idx]; cvec[idx] = SQRTDT * s * s; }
  if (idx < NTRIALS * NN) {
    s0[idx] = 0.5f;                                  // sigmoid(0)
    int m = idx >> 10, nn = idx & 1023;
    out[(long)m * XSEQ_TRIAL_STRIDE + nn] = 0.0f;    // x_seq[:,0,:]
  }
}

// ---------------------------------------------------------------------------
// One 16x16 tile of y = s @ W^T  (K = NN), per wave.
//   s: compact (32 x 1024).  yout: base of output_seq.
// ---------------------------------------------------------------------------
__device__ __forceinline__ void y_tile_wave(const float* __restrict__ s,
                                            const float* __restrict__ Wt,
                                            float* __restrict__ yout, int tcur,
                                            int myt, int oyt) {
  int lane = threadIdx.x & 31;
  int h = lane >> 4, ml = lane & 15;
  int o = oyt * 16 + ml;
  int m0 = myt * 16;
  v8f acc = {0.f, 0.f, 0.f, 0.f, 0.f, 0.f, 0.f, 0.f};
#pragma unroll 8
  for (int k0 = 0; k0 < NN; k0 += 4) {
    int ka = k0 + 2 * h;
    v2f a, b;
    const float* sp = s + (m0 + ml) * NN + ka;
    a[0] = sp[0]; a[1] = sp[1];
    b[0] = Wt[ka * OUTDIM + o];
    b[1] = Wt[(ka + 1) * OUTDIM + o];
    acc = wmma_f32_k4(a, b, acc);
  }
#pragma unroll
  for (int r = 0; r < 8; ++r) {
    int m = m0 + r + 8 * h;
    yout[(long)m * Y_TRIAL_STRIDE + (long)tcur * OUTDIM + o] = acc[r];
  }
}

// ---------------------------------------------------------------------------
// One Euler step. Grid = 17 blocks x 256 threads (8 wave32 per block).
//  Blocks 0..15 : x-update. Block b owns M-tile mt=b/8, N-tiles (b%8)*8+wave.
//                 Stages 16 rows of s_t into 64KB LDS (XOR-swizzled, bank-
//                 conflict-free fragment reads), then K=1024 of f32 WMMA for
//                 s_t @ Jt plus K=64 for u_t @ Bt, then the pointwise update.
//  Block 16     : computes y_{t-1} = s_t @ Wt (waves 0..3), skipped at t==0.
// ---------------------------------------------------------------------------
__global__ __launch_bounds__(256) void rnn_step_kernel(
    const float* __restrict__ inp, const float* __restrict__ Jt,
    const float* __restrict__ Bt, const float* __restrict__ Wt,
    const float* __restrict__ cvec, const float* __restrict__ noise,
    const float* __restrict__ s_in, float* __restrict__ s_out,
    float* __restrict__ out, int t) {
  int blk = blockIdx.x;

  if (blk == 16) {  // readout block: y_{t-1} from s_t
    int wave = threadIdx.x >> 5;
    if (t == 0 || wave >= 4) return;
    y_tile_wave(s_in, Wt, out + XSEQ_TOTAL, t - 1, wave >> 1, wave & 1);
    return;
  }

  __shared__ float lds_s[16 * NN];  // 64KB: rows m=0..15, cols XOR-swizzled by 4*m
  int mt = blk >> 3;                                // 0..1
  int ntile = (blk & 7) * 8 + (threadIdx.x >> 5);   // 0..63

  // Stage s_t rows [mt*16, mt*16+16) into LDS.
  // Loads: 16 lanes read 256B contiguous. Stores: float4, conflict-free per pass.
  {
    int i = threadIdx.x;
    int m = i >> 4;            // 0..15
    int c4 = i & 15;
    const float4* src = (const float4*)(s_in + (mt * 16 + m) * NN);
#pragma unroll
    for (int ch = 0; ch < 16; ++ch) {
      int col4 = ch * 16 + c4;
      float4 v = src[col4];
      int kk = col4 * 4;
      *(float4*)&lds_s[m * NN + (kk ^ (m * 4))] = v;
    }
  }
  __syncthreads();

  int lane = threadIdx.x & 31;
  int h = lane >> 4, ml = lane & 15;
  int n = ntile * 16 + ml;
  int swz = ml * 4;

  v8f acc = {0.f, 0.f, 0.f, 0.f, 0.f, 0.f, 0.f, 0.f};

  // G += sigmoid(x_t) @ J^T   (A from LDS, B rows of Jt are coalesced)
#pragma unroll 8
  for (int k0 = 0; k0 < NN; k0 += 4) {
    int ka = k0 + 2 * h;
    v2f a, b;
    const float* ap = &lds_s[ml * NN + (ka ^ swz)];  // swizzle keeps pairs adjacent
    a[0] = ap[0]; a[1] = ap[1];
    const float* jp = Jt + (long)ka * NN + n;
    b[0] = jp[0]; b[1] = jp[NN];
    acc = wmma_f32_k4(a, b, acc);
  }

  // G += u_t @ B^T   (K = 64)
  {
    const float* up = inp + (long)(mt * 16 + ml) * (TSTEPS * INDIM) + (long)t * INDIM;
#pragma unroll
    for (int k0 = 0; k0 < INDIM; k0 += 4) {
      int ka = k0 + 2 * h;
      v2f a, b;
      a[0] = up[ka]; a[1] = up[ka + 1];
      const float* bp = Bt + (long)ka * NN + n;
      b[0] = bp[0]; b[1] = bp[NN];
      acc = wmma_f32_k4(a, b, acc);
    }
  }

  // x_{t+1} = 0.9 x_t + 0.1 G + c[n]*noise ;  s_{t+1} = sigmoid(x_{t+1})
  float cn = cvec[n];
#pragma unroll
  for (int r = 0; r < 8; ++r) {
    int m = mt * 16 + r + 8 * h;
    long xbase = (long)m * XSEQ_TRIAL_STRIDE + (long)t * NN + n;
    float xold = out[xbase];
    float nz = noise[(long)m * (TSTEPS * NN) + (long)t * NN + n];
    float xnew = (1.0f - DT) * xold + DT * acc[r] + cn * nz;
    out[xbase + NN] = xnew;                 // x_seq[m][t+1][n]
    s_out[m * NN + n] = sigmoidf_(xnew);
  }
}

// Final readout y_{T-1} from s_T. 1 block x 128 threads (4 waves).
__global__ __launch_bounds__(128) void rnn_ylast_kernel(
    const float* __restrict__ s, const float* __restrict__ Wt,
    float* __restrict__ out) {
  int wave = threadIdx.x >> 5;
  y_tile_wave(s, Wt, out + XSEQ_TOTAL, TSTEPS - 1, wave >> 1, wave & 1);
}

// ---------------------------------------------------------------------------
extern "C" void kernel_launch(void* const* d_in, const int* in_sizes, int n_in,
                              void* d_out, int out_size, void* d_ws, size_t ws_size,
                              hipStream_t stream) {
  const float* inp   = (const float*)d_in[0];  // (32,1000,64)
  const float* Bm    = (const float*)d_in[1];  // (1024,64)
  const float* J     = (const float*)d_in[2];  // (1024,1024)
  const float* W     = (const float*)d_in[3];  // (32,1024)
  const float* sig   = (const float*)d_in[4];  // (1024,)
  const float* noise = (const float*)d_in[5];  // (32,1000,1024)
  float* out = (float*)d_out;
  float* ws  = (float*)d_ws;

  // workspace layout (floats): needs ~4.63 MB
  float* Jt   = ws;                       // 1024*1024
  float* Bt   = Jt + NN * NN;             // 64*1024
  float* Wt   = Bt + INDIM * NN;          // 1024*32
  float* cvec = Wt + NN * OUTDIM;         // 1024
  float* sbuf = cvec + NN;                // 2 * 32*1024 (double buffer of sigmoid(x))

  rnn_init_kernel<<<(NN * NN) / 256, 256, 0, stream>>>(Bm, J, W, sig, Jt, Bt, Wt,
                                                       cvec, sbuf, out);
  for (int t = 0; t < TSTEPS; ++t) {
    const float* s_in = sbuf + (t & 1) * (NTRIALS * NN);
    float* s_out      = sbuf + ((t + 1) & 1) * (NTRIALS * NN);
    rnn_step_kernel<<<17, 256, 0, stream>>>(inp, Jt, Bt, Wt, cvec, noise,
                                            s_in, s_out, out, t);
  }
  // s_T lives in buffer (TSTEPS & 1) == 0
  rnn_ylast_kernel<<<1, 128, 0, stream>>>(sbuf + (TSTEPS & 1) * (NTRIALS * NN),
                                          Wt, out);
}